// PPiAttention_21552145891655
// MI455X (gfx1250) — compile-verified
//
#include <hip/hip_runtime.h>
#include <math.h>

#define N_GENES 20000
#define UNITS   10000
#define DEG     32
#define BATCH   128

typedef __attribute__((ext_vector_type(2))) float v2f;
typedef __attribute__((ext_vector_type(8))) float v8f;

// ---------------------------------------------------------------------------
// Kernel 1: transpose feature [BATCH, N_GENES] -> featT [N_GENES, BATCH].
// 32x32 LDS tiles, padded to kill bank conflicts. 20000 = 625*32, 128 = 4*32,
// so no bounds guards needed. Makes every subsequent gene-row gather a fully
// coalesced 512B row (64B per 16-lane half).
// ---------------------------------------------------------------------------
__global__ __launch_bounds__(256) void feat_transpose_kernel(
    const float* __restrict__ in, float* __restrict__ out) {
  __shared__ float tile[32][33];
  const int tx = threadIdx.x;        // 0..31 (gene within tile on load)
  const int ty = threadIdx.y;        // 0..7
  const int g0 = blockIdx.x * 32;
  const int b0 = blockIdx.y * 32;
#pragma unroll
  for (int i = 0; i < 32; i += 8)
    tile[ty + i][tx] = in[(long)(b0 + ty + i) * N_GENES + (g0 + tx)];
  __syncthreads();
#pragma unroll
  for (int i = 0; i < 32; i += 8)
    out[(long)(g0 + ty + i) * BATCH + (b0 + tx)] = tile[tx][ty + i];
}

// ---------------------------------------------------------------------------
// Kernel 2: one block (4 waves) per unit. Wave handles 2 batch-tiles of 16.
// Per 4-deg chunk: each 16-lane half loads 2 contiguous 64B row segments of
// |featT[gene]|, which lands exactly in the 4x16 f32 B-matrix WMMA layout
// (lo half supplies K=0,1; hi half supplies K=2,3). A = ones(16x4) so
// D = sum over K + C, accumulated over 8 chunks => sum over DEG=32 in pure
// f32 via v_wmma_f32_16x16x4_f32. Lane L<16 reads batch (tile*16+L) from
// acc[0] and applies kernel/bias/tanh.
// ---------------------------------------------------------------------------
__global__ __launch_bounds__(128) void ppi_gather_wmma_kernel(
    const float* __restrict__ featT, const int* __restrict__ ppi,
    const float* __restrict__ kernelw, const float* __restrict__ biasw,
    float* __restrict__ out) {
  const int u    = blockIdx.x;
  const int lane = threadIdx.x & 31;
  const int wave = threadIdx.x >> 5;
  const bool hi  = lane >= 16;

  // ppi row is block-uniform -> scalar loads; per-lane K-slot selection.
  int offs[16];
#pragma unroll
  for (int c = 0; c < 8; ++c) {
    const int i0 = ppi[u * DEG + 4 * c + 0];
    const int i1 = ppi[u * DEG + 4 * c + 1];
    const int i2 = ppi[u * DEG + 4 * c + 2];
    const int i3 = ppi[u * DEG + 4 * c + 3];
    offs[2 * c + 0] = (hi ? i2 : i0) * BATCH;
    offs[2 * c + 1] = (hi ? i3 : i1) * BATCH;
  }
  const float kk = kernelw[u];
  const float bb = biasw[u];

  v2f a;
  a.x = 1.0f;
  a.y = 1.0f;

  const int colLane = lane & 15;
#pragma unroll
  for (int t = 0; t < 2; ++t) {
    const int tileN = wave + 4 * t;          // uniform per wave: EXEC all-1s
    const int col   = tileN * 16 + colLane;  // batch column
    v8f acc = {};
#pragma unroll
    for (int c = 0; c < 8; ++c) {
      v2f b;
      b.x = __builtin_fabsf(featT[offs[2 * c + 0] + col]);
      b.y = __builtin_fabsf(featT[offs[2 * c + 1] + col]);
      // D = ones(16x4) x |B|(4x16) + C  => per-column partial deg-sum
      acc = __builtin_amdgcn_wmma_f32_16x16x4_f32(
          false, a, false, b, (short)0, acc, false, false);
    }
    if (lane < 16) {
      const float h = acc[0] * kk + bb;
      out[(long)(tileN * 16 + lane) * UNITS + u] = tanhf(h);
    }
  }
}

extern "C" void kernel_launch(void* const* d_in, const int* in_sizes, int n_in,
                              void* d_out, int out_size, void* d_ws, size_t ws_size,
                              hipStream_t stream) {
  const float* feature = (const float*)d_in[0];   // [128, 20000] f32
  const int*   ppi     = (const int*)d_in[1];     // [10000, 32] i32
  const float* kernelw = (const float*)d_in[2];   // [10000] f32
  const float* biasw   = (const float*)d_in[3];   // [10000] f32
  float*       out     = (float*)d_out;           // [128, 10000] f32
  float*       featT   = (float*)d_ws;            // [20000, 128] f32 = 10.24 MB

  dim3 tBlk(32, 8, 1);
  dim3 tGrd(N_GENES / 32, BATCH / 32, 1);         // 625 x 4
  feat_transpose_kernel<<<tGrd, tBlk, 0, stream>>>(feature, featT);

  ppi_gather_wmma_kernel<<<UNITS, 128, 0, stream>>>(featT, ppi, kernelw, biasw, out);
}